// GCNLayer_75900662055304
// MI455X (gfx1250) — compile-verified
//
#include <hip/hip_runtime.h>
#include <hip/hip_bf16.h>

typedef __attribute__((ext_vector_type(2))) float v2f;
typedef __attribute__((ext_vector_type(8))) float v8f;

#define IN_F  512
#define OUT_F 32
#define KSTEPS (IN_F / 4)            // 128 K-steps of 4
#define WAVES_PER_BLOCK 8

// ---------------------------------------------------------------------------
// Kernel 0: zero the output (harness poisons d_out; atomics need zeros)
// ---------------------------------------------------------------------------
__global__ void zero_f4_kernel(float4* __restrict__ p, int n4) {
    int i = blockIdx.x * blockDim.x + threadIdx.x;
    if (i < n4) p[i] = make_float4(0.f, 0.f, 0.f, 0.f);
}

// ---------------------------------------------------------------------------
// Kernel 1: h[N,32] = x[N,512] @ W[512,32] + bias   via V_WMMA_F32_16X16X4_F32
// One wave per 16-row strip; two 16x16 N-tiles per wave.
// W pre-swizzled in LDS as per-(kstep,tile,lane) float2 B-fragments.
// ---------------------------------------------------------------------------
__global__ void __launch_bounds__(256)
gcn_gemm_wmma_kernel(const float* __restrict__ x,
                     const float* __restrict__ Wg,
                     const float* __restrict__ bias,
                     float* __restrict__ h,
                     int ntiles, int nrows) {
    // 64 KB LDS: sW[(kk*2 + tile)*32 + lane] = {B_vgpr0_val, B_vgpr1_val}
    __shared__ float2 sW[KSTEPS * 2 * 32];

    const int tid  = threadIdx.x;
    const int wave = tid >> 5;
    const int lane = tid & 31;

    // Cooperative swizzle fill: for lane L, tile t, step kk:
    //   VGPR0 value = W[(4kk + 2*(L>>4))*32     + (L&15) + 16t]
    //   VGPR1 value = W[(4kk + 2*(L>>4) + 1)*32 + (L&15) + 16t]
    for (int e = tid; e < KSTEPS * 2 * 32; e += blockDim.x) {
        int kk   = e >> 6;
        int t    = (e >> 5) & 1;
        int L    = e & 31;
        int half = L >> 4;
        int n    = (L & 15) + 16 * t;
        int k0   = 4 * kk + 2 * half;
        sW[e] = make_float2(Wg[k0 * OUT_F + n], Wg[(k0 + 1) * OUT_F + n]);
    }
    __syncthreads();

    const int tile = blockIdx.x * WAVES_PER_BLOCK + wave;
    if (tile >= ntiles) return;

    const int m0   = tile * 16;
    const int half = lane >> 4;
    const int nloc = lane & 15;

    // Accumulators: C layout VGPR v = rows (v, v+8); N = lane&15 (+16 per tile)
    const float b0v = bias[nloc];
    const float b1v = bias[nloc + 16];
    v8f acc0, acc1;
    #pragma unroll
    for (int v = 0; v < 8; ++v) { acc0[v] = b0v; acc1[v] = b1v; }

    // A fragment row for this lane (rows 0..15 mirrored across lane halves)
    const float* __restrict__ arow = x + (long)(m0 + nloc) * IN_F + 2 * half;

    #pragma unroll 4
    for (int kk = 0; kk < KSTEPS; ++kk) {
        float2 a  = *(const float2*)(arow + 4 * kk);
        float2 f0 = sW[(kk * 2 + 0) * 32 + lane];
        float2 f1 = sW[(kk * 2 + 1) * 32 + lane];
        v2f av; av[0] = a.x;  av[1] = a.y;
        v2f bv0; bv0[0] = f0.x; bv0[1] = f0.y;
        v2f bv1; bv1[0] = f1.x; bv1[1] = f1.y;
        acc0 = __builtin_amdgcn_wmma_f32_16x16x4_f32(
                   false, av, false, bv0, (short)0, acc0, false, false);
        acc1 = __builtin_amdgcn_wmma_f32_16x16x4_f32(
                   false, av, false, bv1, (short)0, acc1, false, false);
    }

    // Store: VGPR v -> row m0 + v + 8*half, col (lane&15) + 16*tileN
    #pragma unroll
    for (int v = 0; v < 8; ++v) {
        int row = m0 + v + 8 * half;
        if (row < nrows) {
            float* hp = h + (long)row * OUT_F;
            hp[nloc]      = acc0[v];
            hp[nloc + 16] = acc1[v];
        }
    }
}

// ---------------------------------------------------------------------------
// Kernel 2: edge scatter  out[dst] += ew[e] * h[src]   (f32 atomics, L2-bound)
// 8 threads per edge, float4 per thread.
// ---------------------------------------------------------------------------
__global__ void __launch_bounds__(256)
gcn_scatter_kernel(const float* __restrict__ h,
                   const float* __restrict__ ew,
                   const int* __restrict__ ei,
                   float* __restrict__ out,
                   int E) {
    long long t = (long long)blockIdx.x * blockDim.x + threadIdx.x;
    long long total = (long long)E * 8;
    if (t >= total) return;
    int e = (int)(t >> 3);
    int q = (int)(t & 7);

    int dst = ei[e];        // row 0 of edge_index
    int src = ei[E + e];    // row 1 of edge_index
    float w = ew[e];

    const float4 hv = *(const float4*)(h + (long)src * OUT_F + q * 4);
    float* op = out + (long)dst * OUT_F + q * 4;
    atomicAdd(op + 0, w * hv.x);
    atomicAdd(op + 1, w * hv.y);
    atomicAdd(op + 2, w * hv.z);
    atomicAdd(op + 3, w * hv.w);
}

// ---------------------------------------------------------------------------
extern "C" void kernel_launch(void* const* d_in, const int* in_sizes, int n_in,
                              void* d_out, int out_size, void* d_ws, size_t ws_size,
                              hipStream_t stream) {
    const float* x    = (const float*)d_in[0];   // [N, 512]
    const float* Wg   = (const float*)d_in[1];   // [512, 32]
    const float* bias = (const float*)d_in[2];   // [32]
    const float* ew   = (const float*)d_in[3];   // [E]
    const int*   ei   = (const int*)d_in[4];     // [2, E] (dst row, src row)

    const int N = in_sizes[0] / IN_F;
    const int E = in_sizes[3];

    float* h   = (float*)d_ws;     // [N, 32] intermediate (12.8 MB)
    float* out = (float*)d_out;    // [N, 32]

    // 1) zero output
    int n4 = out_size / 4;
    zero_f4_kernel<<<(n4 + 255) / 256, 256, 0, stream>>>((float4*)out, n4);

    // 2) WMMA GEMM  h = x @ W + b
    int ntiles = (N + 15) / 16;
    int blocks = (ntiles + WAVES_PER_BLOCK - 1) / WAVES_PER_BLOCK;
    gcn_gemm_wmma_kernel<<<blocks, 256, 0, stream>>>(x, Wg, bias, h, ntiles, N);

    // 3) edge scatter with f32 atomics
    long long sthreads = (long long)E * 8;
    int sblocks = (int)((sthreads + 255) / 256);
    gcn_scatter_kernel<<<sblocks, 256, 0, stream>>>(h, ew, ei, out, E);
}